// ExpandedPerformerFeatureMap_41575283425766
// MI455X (gfx1250) — compile-verified
//
#include <hip/hip_runtime.h>

typedef __attribute__((ext_vector_type(2))) float v2f;
typedef __attribute__((ext_vector_type(8))) float v8f;

#define DHEAD 64
#define NFEAT 256
#define TILE_M 16
#define TILES_PER_BLOCK 4
#define THREADS 512   // 16 wave32 waves; wave j owns output cols [16j, 16j+16)

__global__ __launch_bounds__(THREADS)
void performer_featmap_wmma(const float* __restrict__ x,
                            const float* __restrict__ w,
                            float* __restrict__ out,
                            int rows) {
    const int lane = threadIdx.x & 31;
    const int wave = threadIdx.x >> 5;   // j-tile index 0..15
    const int l0   = lane & 15;
    const int hi   = lane >> 4;

    // ---- B fragments, loaded ONCE per wave and held in registers.
    // WMMA 16x16x4 f32 B (4x16, KxN) layout: VGPR v holds K = v + 2*hi, lane n = N.
    // B[k][n] = s * W[16*j + n][k]  with s = 64^-0.25 folded in here.
    const float s = 0.35355339059327373f;   // 64^(-1/4) = 2^(-1.5)
    v2f bfrag[16];
    {
        const float* wp = w + (size_t)(wave * 16 + l0) * DHEAD + 2 * hi;
        #pragma unroll
        for (int k = 0; k < 16; ++k) {
            v2f t = *(const v2f*)(wp + 4 * k);
            bfrag[k].x = t.x * s;
            bfrag[k].y = t.y * s;
        }
    }

    for (int t = 0; t < TILES_PER_BLOCK; ++t) {
        const int rowBase = (blockIdx.x * TILES_PER_BLOCK + t) * TILE_M;
        if (rowBase >= rows) break;   // block-uniform: EXEC stays all-ones

        // ---- A fragments (16 rows x K=64) + per-row sum of squares.
        // A (16x4, MxK) layout: VGPR v holds K = v + 2*hi, lane m = M.
        v2f afrag[16];
        float nss = 0.f;
        const float* xp = x + (size_t)(rowBase + l0) * DHEAD + 2 * hi;
        #pragma unroll
        for (int k = 0; k < 16; ++k) {
            v2f a = *(const v2f*)(xp + 4 * k);
            afrag[k] = a;
            nss += a.x * a.x + a.y * a.y;
        }
        // lane(l0,hi) covered K = {2hi, 2hi+1} mod 4; combine both halves:
        nss += __shfl_xor(nss, 16, 32);      // lane r (and r+16) now holds ||x_row(r)||^2

        // coef for the 8 output rows this lane writes (row = g + 8*hi):
        // hx/sqrt(m) = exp(-0.5 * (1/8) * ||x||^2) / 16 = exp(-||x||^2/16)/16
        float coef[8];
        #pragma unroll
        for (int g = 0; g < 8; ++g) {
            float ns = __shfl(nss, g + 8 * hi, 32);
            coef[g] = __expf(-0.0625f * ns) * 0.0625f;
        }

        // ---- GEMM: 16 chained V_WMMA_F32_16X16X4_F32 cover K=64.
        v8f acc = {};
        #pragma unroll
        for (int k = 0; k < 16; ++k) {
            acc = __builtin_amdgcn_wmma_f32_16x16x4_f32(
                false, afrag[k], false, bfrag[k],
                (short)0, acc, false, false);
        }

        // ---- epilogue: D VGPR g -> row rowBase + g + 8*hi, col = 16*wave + l0.
        // Output is write-once: non-temporal stores keep W / x tiles in cache.
        float* op = out + (size_t)rowBase * NFEAT + wave * 16 + l0;
        #pragma unroll
        for (int g = 0; g < 8; ++g) {
            float v = coef[g] * __expf(acc[g]);
            __builtin_nontemporal_store(v, op + (size_t)(g + 8 * hi) * NFEAT);
        }
    }
}

extern "C" void kernel_launch(void* const* d_in, const int* in_sizes, int n_in,
                              void* d_out, int out_size, void* d_ws, size_t ws_size,
                              hipStream_t stream) {
    const float* x = (const float*)d_in[0];          // (4,16,4096,64) fp32
    const float* w = (const float*)d_in[1];          // (256,64) fp32
    float* out = (float*)d_out;                      // (4,16,4096,256) fp32

    const int rows   = in_sizes[0] / DHEAD;                       // 262144
    const int tiles  = (rows + TILE_M - 1) / TILE_M;              // 16384
    const int blocks = (tiles + TILES_PER_BLOCK - 1) / TILES_PER_BLOCK; // 4096

    performer_featmap_wmma<<<blocks, THREADS, 0, stream>>>(x, w, out, rows);
}